// CASSI_74775380623701
// MI455X (gfx1250) — compile-verified
//
#include <hip/hip_runtime.h>
#include <stdint.h>

// ---------------------------------------------------------------------------
// CASSI forward (mask -> dispersed scatter-add -> shifted gather -> mask)
// MI455X / gfx1250. Memory-bound (~470 MB traffic, ~0.18 GFLOP => ~0.4 FLOP/B;
// HBM floor ~20us @ 23.3 TB/s). No matmul structure => no WMMA; instead use
// the CDNA5 Tensor Data Mover to DMA the 28-row strided spectral tile into
// LDS asynchronously (double-buffered, TENSORcnt-tracked), and stream the
// output with non-temporal b128 stores.
// ---------------------------------------------------------------------------

#define B_      8
#define L_      28
#define M_      512
#define N_      512
#define NOUT    566          // N + STRIDE*(L-1) = 512 + 2*27
#define TASKS   8            // (b,i) rows per workgroup
#define WGS     512          // WGS * TASKS == B_*M_ == 4096
#define THREADS 256

typedef uint32_t u32x4 __attribute__((ext_vector_type(4)));
typedef uint32_t u32x8 __attribute__((ext_vector_type(8)));
typedef float    f32x2 __attribute__((ext_vector_type(2)));
typedef float    f32x4 __attribute__((ext_vector_type(4)));

// Issue one TDM 2-D tile load: 28 rows x 512 f32, row stride M*N elements,
// global -> LDS. D# group0 (4 SGPRs) + group1 (8 SGPRs); groups 2/3 NULL
// (<=2D tensor). Tracked by TENSORcnt.
__device__ __forceinline__ void tdm_issue_tile(const float* gsrc, unsigned lds_off) {
    uint64_t ga = (uint64_t)(uintptr_t)gsrc;
    u32x4 g0;
    g0.x = 1u;                                   // count=1 valid descriptor, user mode
    g0.y = lds_off;                              // lds_addr (bytes)
    g0.z = (uint32_t)ga;                         // global_addr[31:0]
    g0.w = (uint32_t)((ga >> 32) & 0x01FFFFFFu)  // global_addr[56:32]
         | (2u << 30);                           // type=2 ("image")
    u32x8 g1;
    g1.s0 = (2u << 16);                          // data_size=2 (4 bytes); no multicast/pad
    g1.s1 = ((uint32_t)N_ & 0xFFFFu) << 16;      // tensor_dim0[15:0]=512
    g1.s2 = ((uint32_t)L_ & 0xFFFFu) << 16;      // tensor_dim0 hi=0 | tensor_dim1[15:0]=28
    g1.s3 = ((uint32_t)N_) << 16;                // tensor_dim1 hi=0 | tile_dim0=512
    g1.s4 = (uint32_t)L_;                        // tile_dim1=28 | tile_dim2=0
    g1.s5 = (uint32_t)(M_ * N_);                 // tensor_dim0_stride lo32 = 262144 elems
    g1.s6 = 0u;                                  // stride hi | tensor_dim1_stride lo (unused)
    g1.s7 = 0u;
    asm volatile("tensor_load_to_lds %0, %1" :: "s"(g0), "s"(g1) : "memory");
}

__global__ __launch_bounds__(THREADS)
void cassi_fwd_tdm_kernel(const float* __restrict__ x,
                          const float* __restrict__ phi,
                          float* __restrict__ out) {
    __shared__ float xbuf[2][L_ * N_];       // 2 x 56 KB double-buffered TDM tiles
    __shared__ float phb[N_];                // phi row
    __shared__ float y2[NOUT + 2];           // dispersed accumulator (pad for f32x2)

    const unsigned tid  = threadIdx.x;
    // readfirstlane => compiler-visible uniform wave id => scalar branch.
    // (TDM ignores EXEC, so an exec-masked path could issue duplicate DMAs.)
    const unsigned wid  = ((unsigned)__builtin_amdgcn_readfirstlane((int)tid)) >> 5;
    const unsigned row0 = blockIdx.x * TASKS;

    const unsigned lds0 = (unsigned)(uintptr_t)&xbuf[0][0];
    const unsigned lds1 = (unsigned)(uintptr_t)&xbuf[1][0];

    // Prologue: DMA task 0's tile into buffer 0.
    if (wid == 0) {
        const unsigned r = row0;
        const unsigned b = r >> 9, i = r & 511u;
        tdm_issue_tile(x + ((size_t)b * L_ * M_ + i) * N_, lds0);
    }

    for (int k = 0; k < TASKS; ++k) {
        const unsigned r = row0 + (unsigned)k;
        const unsigned b = r >> 9, i = r & 511u;

        if (wid == 0) {
            if (k + 1 < TASKS) {
                const unsigned rn = r + 1u;
                const unsigned bn = rn >> 9, in_ = rn & 511u;
                // prefetch next tile into the other buffer, then wait for
                // the *oldest* TDM op (current tile): in-order => cnt<=1.
                tdm_issue_tile(x + ((size_t)bn * L_ * M_ + in_) * N_,
                               (k & 1) ? lds0 : lds1);
                __builtin_amdgcn_s_wait_tensorcnt(1);
            } else {
                __builtin_amdgcn_s_wait_tensorcnt(0);
            }
        }

        // Stage phi row (coalesced 2 KB).
        const float* prow = phi + (size_t)i * N_;
        phb[tid]        = prow[tid];
        phb[tid + 256]  = prow[tid + 256];
        __syncthreads();   // TDM tile + phi visible to all waves

        const float* xb = xbuf[k & 1];

        // y2[c] = sum_l x[l][c-2l] * phi[c-2l]   (banded shift-reduce)
        for (unsigned c = tid; c < NOUT; c += THREADS) {
            int lmin = (c > (N_ - 1)) ? (int)((c - (unsigned)(N_ - 2)) >> 1) : 0; // ceil((c-511)/2)
            int lmax = (int)(c >> 1); if (lmax > L_ - 1) lmax = L_ - 1;
            float acc = 0.0f;
            for (int l = lmin; l <= lmax; ++l) {
                const int kk = (int)c - 2 * l;
                acc += xb[l * N_ + kk] * phb[kk];   // conflict-free: consecutive lanes,
            }                                        // consecutive banks
            y2[c] = acc;
        }
        __syncthreads();

        // out[b,l,i,j] = phi[j] * y2[2l+j]; streamed 16B non-temporal stores.
        float* ob = out + ((size_t)b * L_ * M_ + i) * N_;   // &out[b,0,i,0]
        #pragma unroll 2
        for (unsigned q = tid; q < (L_ * N_) / 4; q += THREADS) {
            const unsigned l  = q >> 7;              // q / (N/4)
            const unsigned j0 = (q & 127u) << 2;
            const unsigned c0 = 2u * l + j0;         // even => 8B-aligned LDS reads
            const f32x2 ya = *(const f32x2*)&y2[c0];
            const f32x2 yb = *(const f32x2*)&y2[c0 + 2];
            const f32x4 p4 = *(const f32x4*)&phb[j0];
            f32x4 o;
            o.x = ya.x * p4.x;
            o.y = ya.y * p4.y;
            o.z = yb.x * p4.z;
            o.w = yb.y * p4.w;
            __builtin_nontemporal_store(o, (f32x4*)(ob + (size_t)l * M_ * N_ + j0));
        }
        __syncthreads();   // protect xbuf[(k)&1]^... reuse and phb overwrite
    }
}

extern "C" void kernel_launch(void* const* d_in, const int* in_sizes, int n_in,
                              void* d_out, int out_size, void* d_ws, size_t ws_size,
                              hipStream_t stream) {
    const float* x   = (const float*)d_in[0];   // (B, L, M, N) f32
    const float* phi = (const float*)d_in[1];   // (M, N) f32
    float* out = (float*)d_out;                 // (B, L, M, N) f32
    (void)in_sizes; (void)n_in; (void)out_size; (void)d_ws; (void)ws_size;
    cassi_fwd_tdm_kernel<<<dim3(WGS), dim3(THREADS), 0, stream>>>(x, phi, out);
}